// CombinedLoss_56684978372920
// MI455X (gfx1250) — compile-verified
//
#include <hip/hip_runtime.h>
#include <stdint.h>

// ---------------------------------------------------------------------------
// Problem geometry (fixed by the reference's setup_inputs)
// ---------------------------------------------------------------------------
constexpr int NB = 2;            // batch
constexpr int CH = 2;            // channels in net_output
constexpr int DD = 64;           // depth
constexpr int HH = 128;          // height
constexpr int WW = 128;          // width
constexpr long long S   = (long long)DD * HH * WW;   // 1,048,576 voxels / batch
constexpr long long NVT = (long long)NB * S;         // 2,097,152 voxels total
constexpr int NROWS     = NB * DD * HH;              // 16,384 x-rows
constexpr int PW_BLOCKS = (int)(NVT / 256);          // 8,192 pointwise blocks

// Scalar accumulator layout (floats, some slots used as int bit-patterns)
//  0 tp1  1 fp1  2 fn1  3 ce  4 conn  5 dirsum
//  6 sp*yt  7 sp  8 st*p  9 st
//  10..11 rmaxL[b] 12..13 rminL[b] 14..15 rmaxP[b] 16..17 rminP[b]
//  18 inter1 19 union1 20 inter2 21 union2
constexpr int NACC = 32;

// ---------------------------------------------------------------------------
// CDNA5 feature gates
// ---------------------------------------------------------------------------
#if defined(__gfx1250__) && defined(__has_builtin)
#  if __has_builtin(__builtin_amdgcn_tensor_load_to_lds)
#    define CDNA5_TDM 1
#  endif
#endif
#ifndef CDNA5_TDM
#  define CDNA5_TDM 0
#endif

#if defined(__gfx1250__)
#  if __has_builtin(__builtin_amdgcn_s_wait_tensorcnt)
#    define TDM_WAIT() __builtin_amdgcn_s_wait_tensorcnt(0)
#  else
#    define TDM_WAIT() asm volatile("s_wait_tensorcnt 0x0" ::: "memory")
#  endif
#else
#  define TDM_WAIT()
#endif

typedef float v2f __attribute__((ext_vector_type(2)));
typedef float v8f __attribute__((ext_vector_type(8)));

// ---------------------------------------------------------------------------
// Wave reduction via WMMA: D = A(16x4) * ones(4x16) + C.
// Lane L contributes s at A[L%16][L<16?0:2]; each D column holds row sums, so
// sum(c[0..7]) on lanes 0-15 covers rows 0-7 (s_m + s_{m+16}), lanes 16-31
// cover rows 8-15; one xor-16 shuffle completes the 32-lane total.
// ---------------------------------------------------------------------------
__device__ __forceinline__ float wave_sum(float v) {
#if defined(__gfx1250__) && __has_builtin(__builtin_amdgcn_wmma_f32_16x16x4_f32)
  v2f a; a.x = v;    a.y = 0.0f;
  v2f b; b.x = 1.0f; b.y = 1.0f;
  v8f c = {0.f,0.f,0.f,0.f,0.f,0.f,0.f,0.f};
  c = __builtin_amdgcn_wmma_f32_16x16x4_f32(false, a, false, b, (short)0, c,
                                            false, false);
  float s = c[0]+c[1]+c[2]+c[3]+c[4]+c[5]+c[6]+c[7];
  s += __shfl_xor(s, 16, 32);
  return s;
#else
  for (int o = 16; o > 0; o >>= 1) v += __shfl_xor(v, o, 32);
  return v;
#endif
}

__device__ __forceinline__ void block_atomic_add(float v, float* acc) {
  float s = wave_sum(v);
  if ((threadIdx.x & 31u) == 0u) atomicAdd(acc, s);
}

__device__ __forceinline__ float wave_max(float v) {
  for (int o = 16; o > 0; o >>= 1) v = fmaxf(v, __shfl_xor(v, o, 32));
  return v;
}
__device__ __forceinline__ float wave_min(float v) {
  for (int o = 16; o > 0; o >>= 1) v = fminf(v, __shfl_xor(v, o, 32));
  return v;
}

// ---------------------------------------------------------------------------
// TDM gather-mode row staging: one tensor_load_to_lds pulls up to 9 clamped
// halo rows (16-bit row indices into a [tensor_rows x WW] fp32 tensor) into
// LDS. amdgpu-toolchain (clang-23) 6-arg builtin form:
//   (uint32x4 g0, int32x8 g1, int32x4 g2, int32x4 g3, int32x8 gx, i32 cpol)
// ---------------------------------------------------------------------------
#if CDNA5_TDM
typedef unsigned int v4u __attribute__((ext_vector_type(4)));
typedef int          v8i __attribute__((ext_vector_type(8)));
typedef int          v4i __attribute__((ext_vector_type(4)));

__device__ __forceinline__ void tdm_gather_rows(const void* base,
                                                unsigned lds_addr,
                                                const int* rows, int nrows,
                                                unsigned tensor_rows) {
  unsigned long long ga = (unsigned long long)base;
  v4u g0;
  g0[0] = 0x80000001u;                                  // gather_mode, 16b idx, count=1
  g0[1] = lds_addr;                                     // LDS byte address
  g0[2] = (unsigned)ga;                                 // global_addr[31:0]
  g0[3] = (unsigned)((ga >> 32) & 0x01FFFFFFull) | (2u << 30);  // addr hi | type=2
  unsigned dim0 = (unsigned)WW;                         // tensor_dim0 (elements)
  unsigned dim1 = tensor_rows;                          // tensor_dim1 (rows)
  v8i g1;
  g1[0] = (int)(2u << 16);                              // data_size = 4B
  g1[1] = (int)((dim0 & 0xFFFFu) << 16);                // tensor_dim0 lo16
  g1[2] = (int)(((dim0 >> 16) & 0xFFFFu) | ((dim1 & 0xFFFFu) << 16));
  g1[3] = (int)(((dim1 >> 16) & 0xFFFFu) | ((unsigned)WW << 16)); // tile_dim0
  g1[4] = nrows;                                        // tile_dim1 = #indices
  g1[5] = (int)WW;                                      // tensor_dim0_stride lo32
  g1[6] = 0;
  g1[7] = 0;
  unsigned p[5] = {0u,0u,0u,0u,0u};
  for (int i = 0; i < nrows; ++i)
    p[i >> 1] |= ((unsigned)rows[i] & 0xFFFFu) << ((i & 1) * 16);
  v4i g2, g3;
  g2[0]=(int)p[0]; g2[1]=(int)p[1]; g2[2]=(int)p[2]; g2[3]=(int)p[3];
  g3[0]=(int)p[4]; g3[1]=0; g3[2]=0; g3[3]=0;
  v8i gx = {0,0,0,0,0,0,0,0};                           // extra group (clang-23 form)
  __builtin_amdgcn_tensor_load_to_lds(g0, g1, g2, g3, gx, 0);
}
#endif

// Stage nrows rows (4-byte elements) into LDS tile. Wave0 issues one TDM
// gather; everyone barriers. Fallback: cooperative vector loads.
__device__ __forceinline__ void stage_rows(void* tile, const void* base,
                                           const int* rows, int nrows,
                                           unsigned tensor_rows) {
#if CDNA5_TDM
  if (threadIdx.x < 32u) {   // wave-uniform branch: wave 0 drives the TDM
    tdm_gather_rows(base, (unsigned)(unsigned long long)tile, rows, nrows,
                    tensor_rows);
    TDM_WAIT();
  }
  __syncthreads();
#else
  const unsigned* b = (const unsigned*)base;
  unsigned* t = (unsigned*)tile;
  for (int i = 0; i < nrows; ++i)
    t[i * WW + threadIdx.x] = b[(size_t)rows[i] * WW + threadIdx.x];
  __syncthreads();
#endif
}

// ---------------------------------------------------------------------------
// Kernels
// ---------------------------------------------------------------------------

__global__ void k_init_acc(float* accF) {
  int t = threadIdx.x;
  if (t < NACC) {
    bool mn = (t == 12 || t == 13 || t == 16 || t == 17);
    accF[t] = mn ? __int_as_float(0x7F800000) : 0.0f;   // +inf for min slots
  }
}

__global__ void k_fill(float* p, float v) {
  long long i = (long long)blockIdx.x * blockDim.x + threadIdx.x;
  if (i < NVT) p[i] = v;
}

// prob/ytrue/hard + dice(tp1,fp1,fn1) + CE + connectivity sums
__global__ void k_prep(const float* __restrict__ net, const int* __restrict__ tgt,
                       float* prob, float* ytrue, float* hard, float* accF) {
  long long i = (long long)blockIdx.x * blockDim.x + threadIdx.x;
  float tp1 = 0.f, fp1 = 0.f, fn1 = 0.f, ce = 0.f, conn = 0.f;
  if (i < NVT) {
    long long n = i / S, r = i % S;
    float x0 = net[(n * CH + 0) * S + r];
    float x1 = net[(n * CH + 1) * S + r];
    int   t  = tgt[i];
    float y  = (t > 0) ? 1.f : 0.f;
    float p1 = 1.f / (1.f + expf(x0 - x1));           // softmax([x0,x1])[1]
    float m  = fmaxf(x0, x1);
    float lse = m + logf(expf(x0 - m) + expf(x1 - m));
    ce  = lse - ((t > 0) ? x1 : x0);                  // -log p_t
    tp1 = p1 * y;
    fp1 = p1 * (1.f - y);
    fn1 = (1.f - p1) * y;
    float tb = y;                                     // float(t) > 0.5
    conn = fabsf(((x0 > 0.5f) ? 1.f : 0.f) - tb) +
           fabsf(((x1 > 0.5f) ? 1.f : 0.f) - tb);
    prob[i]  = p1;
    ytrue[i] = y;
    hard[i]  = (p1 > 0.5f) ? 1.f : 0.f;
  }
  block_atomic_add(tp1,  accF + 0);
  block_atomic_add(fp1,  accF + 1);
  block_atomic_add(fn1,  accF + 2);
  block_atomic_add(ce,   accF + 3);
  block_atomic_add(conn, accF + 4);
}

// soft_erode: min over the 7-point cross (border clamp == window shrink)
__global__ void k_erode(const float* __restrict__ src, float* __restrict__ dst) {
  __shared__ float tile[5 * WW];
  int bid = blockIdx.x;
  int n = bid / (DD * HH); int rem = bid % (DD * HH);
  int z = rem / HH;        int y   = rem % HH;
  int zm = max(z - 1, 0), zp = min(z + 1, DD - 1);
  int ym = max(y - 1, 0), yp = min(y + 1, HH - 1);
  int rows[5] = { zm * HH + y, z * HH + ym, z * HH + y, z * HH + yp, zp * HH + y };
  stage_rows(tile, src + (size_t)n * S, rows, 5, (unsigned)(DD * HH));
  int x  = threadIdx.x;
  int xm = max(x - 1, 0), xp = min(x + 1, WW - 1);
  float m = tile[2 * WW + x];
  m = fminf(m, tile[2 * WW + xm]);
  m = fminf(m, tile[2 * WW + xp]);
  m = fminf(m, tile[1 * WW + x]);
  m = fminf(m, tile[3 * WW + x]);
  m = fminf(m, tile[0 * WW + x]);
  m = fminf(m, tile[4 * WW + x]);
  dst[(size_t)bid * WW + x] = m;
}

// 3x3x3 cube min/max pool; optional fused EDT accumulate (acc += src center)
__global__ void k_pool_cube(const float* __restrict__ src, float* __restrict__ dst,
                            float* acc, int isMin) {
  __shared__ float tile[9 * WW];
  int bid = blockIdx.x;
  int n = bid / (DD * HH); int rem = bid % (DD * HH);
  int z = rem / HH;        int y   = rem % HH;
  int rows[9]; int q = 0;
  for (int dz = -1; dz <= 1; ++dz) {
    int zz = min(max(z + dz, 0), DD - 1);
    for (int dy = -1; dy <= 1; ++dy) {
      int yy = min(max(y + dy, 0), HH - 1);
      rows[q++] = zz * HH + yy;
    }
  }
  stage_rows(tile, src + (size_t)n * S, rows, 9, (unsigned)(DD * HH));
  int x = threadIdx.x;
  float center = tile[4 * WW + x];
  float m = center;
  for (int r = 0; r < 9; ++r)
    for (int dx = -1; dx <= 1; ++dx) {
      int xx = min(max(x + dx, 0), WW - 1);
      float v = tile[r * WW + xx];
      m = isMin ? fminf(m, v) : fmaxf(m, v);
    }
  size_t o = (size_t)bid * WW + x;
  dst[o] = m;
  if (acc) acc[o] += center;
}

__global__ void k_skel_init(const float* __restrict__ img,
                            const float* __restrict__ open_, float* skel) {
  long long i = (long long)blockIdx.x * blockDim.x + threadIdx.x;
  if (i >= NVT) return;
  float d = img[i] - open_[i];
  skel[i] = d > 0.f ? d : 0.f;
}

__global__ void k_skel_update(const float* __restrict__ img,
                              const float* __restrict__ open_, float* skel) {
  long long i = (long long)blockIdx.x * blockDim.x + threadIdx.x;
  if (i >= NVT) return;
  float d = img[i] - open_[i];
  float delta = d > 0.f ? d : 0.f;
  float s = skel[i];
  float t = delta - s * delta;
  skel[i] = s + (t > 0.f ? t : 0.f);
}

// sum(a*b) -> accF[idot], sum(a) -> accF[isum]
__global__ void k_dot2(const float* __restrict__ a, const float* __restrict__ b,
                       float* accF, int idot, int isum) {
  long long i = (long long)blockIdx.x * blockDim.x + threadIdx.x;
  float va = 0.f, prod = 0.f;
  if (i < NVT) { va = a[i]; prod = va * b[i]; }
  block_atomic_add(prod, accF + idot);
  block_atomic_add(va,   accF + isum);
}

// Directional (Sobel cosine) loss: gx = ss[w]*cc[h], gy = ss[w]*cc[d],
// gz = ss[d]*cc[w] (faithful to the reference's transposes), zero padding.
__global__ void k_sobel(const float* __restrict__ net, const int* __restrict__ tgt,
                        float* accF) {
  __shared__ float tp[9 * WW];
  __shared__ int   tt[9 * WW];
  int bid = blockIdx.x;
  int n = bid / (DD * HH); int rem = bid % (DD * HH);
  int z = rem / HH;        int y   = rem % HH;
  int rows[9]; int q = 0;
  for (int dz = -1; dz <= 1; ++dz) {
    int zz = min(max(z + dz, 0), DD - 1);
    for (int dy = -1; dy <= 1; ++dy) {
      int yy = min(max(y + dy, 0), HH - 1);
      rows[q++] = zz * HH + yy;
    }
  }
  stage_rows(tp, net + (size_t)(n * CH) * S, rows, 9, (unsigned)(DD * HH));
  stage_rows(tt, tgt + (size_t)n * S,        rows, 9, (unsigned)(DD * HH));
  int x = threadIdx.x;
  float gpx=0.f,gpy=0.f,gpz=0.f, gtx=0.f,gty=0.f,gtz=0.f;
  for (int dz = -1; dz <= 1; ++dz) {
    if (z + dz < 0 || z + dz >= DD) continue;
    for (int dy = -1; dy <= 1; ++dy) {
      if (y + dy < 0 || y + dy >= HH) continue;
      int r = (dz + 1) * 3 + (dy + 1);
      for (int dx = -1; dx <= 1; ++dx) {
        if (x + dx < 0 || x + dx >= WW) continue;
        float vp = tp[r * WW + x + dx];
        float vt = (float)tt[r * WW + x + dx];
        float sw = (float)dx, sd = (float)dz;
        float ch = 2.f - (float)(dy * dy);
        float cd = 2.f - (float)(dz * dz);
        float cw = 2.f - (float)(dx * dx);
        gpx += sw * ch * vp; gpy += sw * cd * vp; gpz += sd * cw * vp;
        gtx += sw * ch * vt; gty += sw * cd * vt; gtz += sd * cw * vt;
      }
    }
  }
  float np_ = sqrtf(gpx*gpx + gpy*gpy + gpz*gpz);
  float dp  = fmaxf(np_, 1e-12f);
  float ax = gpx/dp, ay = gpy/dp, az = gpz/dp;
  float nt_ = sqrtf(gtx*gtx + gty*gty + gtz*gtz);
  float dt  = fmaxf(nt_, 1e-12f);
  float bx = gtx/dt, by = gty/dt, bz = gtz/dt;
  float num = ax*bx + ay*by + az*bz;
  float na  = sqrtf(ax*ax + ay*ay + az*az);
  float nb  = sqrtf(bx*bx + by*by + bz*bz);
  float den = fmaxf(na * nb, 1e-8f);
  block_atomic_add(num / den, accF + 5);
}

// Per-batch skel_radius max/min (bit-cast integer atomics: exact + deterministic)
__global__ void k_radius(const float* __restrict__ accT, const float* __restrict__ skelT,
                         const float* __restrict__ ytrue,
                         const float* __restrict__ accH, const float* __restrict__ skelH,
                         const float* __restrict__ prob, const float* __restrict__ hard,
                         int* accI) {
  long long i = (long long)blockIdx.x * blockDim.x + threadIdx.x;
  float rl = 0.f, rp = 0.f;
  int n = 0;
  if (i < NVT) {
    n = (int)(i / S);
    rl = (accT[i] * ytrue[i]) * skelT[i];
    float spp = skelH[i] * prob[i];
    float sb  = spp > 0.5f ? 1.f : 0.f;
    rp = (accH[i] * hard[i]) * sb;
  }
  float wl_max = wave_max(rl), wl_min = wave_min(rl);
  float wp_max = wave_max(rp), wp_min = wave_min(rp);
  if ((threadIdx.x & 31u) == 0u) {   // n uniform within a wave (rows align)
    atomicMax(&accI[10 + n], __float_as_int(wl_max));
    atomicMin(&accI[12 + n], __float_as_int(wl_min));
    atomicMax(&accI[14 + n], __float_as_int(wp_max));
    atomicMin(&accI[16 + n], __float_as_int(wp_min));
  }
}

// Union-loss weighted sums (two general_union terms)
__global__ void k_union(const float* __restrict__ accT, const float* __restrict__ skelT,
                        const float* __restrict__ ytrue,
                        const float* __restrict__ accH, const float* __restrict__ skelH,
                        const float* __restrict__ prob, const float* __restrict__ hard,
                        const int* __restrict__ accI, float* accF) {
  long long i = (long long)blockIdx.x * blockDim.x + threadIdx.x;
  float i1 = 0.f, u1 = 0.f, i2 = 0.f, u2 = 0.f;
  if (i < NVT) {
    int n = (int)(i / S);
    float rmaxL = fmaxf(__int_as_float(accI[10 + n]), 1.f);
    float rminL = fmaxf(__int_as_float(accI[12 + n]), 1.f);
    float rmaxP = fmaxf(__int_as_float(accI[14 + n]), 1.f);
    float rminP = fmaxf(__int_as_float(accI[16 + n]), 1.f);
    // label side (prob_flag = False)
    float y  = ytrue[i], sT = skelT[i];
    float dL = accT[i] * y;
    float q_vl = (fminf(dL, rmaxL) / rmaxL) * y;
    float srL  = dL * sT;
    float tL   = (rmaxL - srL + rminL) / rmaxL;
    float q_sl = tL * tL * sT * sT;                  // I_norm * skel
    // pred side (prob_flag = True)
    float h = hard[i], p = prob[i];
    float s_in = skelH[i] * p;                       // skel_pred_prob
    float sb   = s_in > 0.5f ? 1.f : 0.f;
    float dP   = accH[i] * h;
    float q_vp = (fminf(dP, rmaxP) / rmaxP) * p;
    float srP  = dP * sb;
    float tP   = (rmaxP - srP + rminP) / rmaxP;
    float q_sp = tP * tP * sb * s_in;                // I_norm * skel_input
    i1 = q_sp * powf(q_sp + 1e-4f, 0.7f) * q_vl;
    u1 = q_sp * (0.1f * q_sp + 0.9f * q_vl);
    i2 = q_sl * powf(q_vp + 1e-4f, 0.7f) * q_sl;
    u2 = q_sl * (0.1f * q_vp + 0.9f * q_sl);
  }
  block_atomic_add(i1, accF + 18);
  block_atomic_add(u1, accF + 19);
  block_atomic_add(i2, accF + 20);
  block_atomic_add(u2, accF + 21);
}

__global__ void k_final(const float* __restrict__ accF, float* out) {
  if (threadIdx.x == 0 && blockIdx.x == 0) {
    float tp1 = accF[0], fp1 = accF[1], fn1 = accF[2];
    float dice = -((2.f * tp1 + 1e-5f) / (2.f * tp1 + fp1 + fn1 + 1e-5f));
    float ce   = accF[3] / (float)NVT;
    float conn = accF[4] / (float)(2 * NVT);
    float dir  = 1.f - accF[5] / (float)NVT;
    float tprec = (accF[6] + 1.f) / (accF[7] + 1.f);
    float tsens = (accF[8] + 1.f) / (accF[9] + 1.f);
    float cl    = 1.f - 2.f * tprec * tsens / (tprec + tsens);
    float L1 = 1.f - (accF[18] + 1.f) / (accF[19] + 1.f);
    float L2 = 1.f - (accF[20] + 1.f) / (accF[21] + 1.f);
    out[0] = dice + ce + cl + dir + conn + L1 + L2;
  }
}

// ---------------------------------------------------------------------------
// Host orchestration
// ---------------------------------------------------------------------------
static void soft_skel_seq(const float* X, float* SK, float* A, float* B,
                          float* C, hipStream_t stream) {
  // m_{k+1} = erode(m_k); open(m_k) = dilate(m_{k+1}) -> share the erode chain
  dim3 gs(NROWS), bs(WW);
  dim3 gp(PW_BLOCKS), bp(256);
  k_erode    <<<gs, bs, 0, stream>>>(X, A);                 // m1
  k_pool_cube<<<gs, bs, 0, stream>>>(A, B, nullptr, 0);     // dilate(m1)
  k_skel_init<<<gp, bp, 0, stream>>>(X, B, SK);
  float* cur = A; float* nxt = B;
  for (int it = 0; it < 10; ++it) {
    k_erode      <<<gs, bs, 0, stream>>>(cur, nxt);             // m_{k+1}
    k_pool_cube  <<<gs, bs, 0, stream>>>(nxt, C, nullptr, 0);   // open(m_k)
    k_skel_update<<<gp, bp, 0, stream>>>(cur, C, SK);
    float* t = cur; cur = nxt; nxt = t;
  }
}

static void edt_seq(const float* mask, float* acc, float* A, float* B,
                    hipStream_t stream) {
  dim3 gs(NROWS), bs(WW);
  dim3 gp(PW_BLOCKS), bp(256);
  k_fill<<<gp, bp, 0, stream>>>(acc, 0.f);
  const float* src = mask; float* pA = A; float* pB = B;
  for (int j = 0; j < 16; ++j) {          // acc += cur; cur = minpool3(cur)
    k_pool_cube<<<gs, bs, 0, stream>>>(src, pA, acc, 1);
    src = pA; float* t = pA; pA = pB; pB = t;
  }
}

extern "C" void kernel_launch(void* const* d_in, const int* in_sizes, int n_in,
                              void* d_out, int out_size, void* d_ws, size_t ws_size,
                              hipStream_t stream) {
  const float* net = (const float*)d_in[0];   // (2,2,64,128,128) f32
  const int*   tgt = (const int*)d_in[1];     // (2,1,64,128,128) i32
  float* out = (float*)d_out;

  float* ws    = (float*)d_ws;                // needs ~84 MB of scratch
  float* prob  = ws + 0LL * NVT;
  float* ytrue = ws + 1LL * NVT;
  float* hard  = ws + 2LL * NVT;
  float* A     = ws + 3LL * NVT;
  float* B     = ws + 4LL * NVT;
  float* C     = ws + 5LL * NVT;
  float* SKT   = ws + 6LL * NVT;              // soft_skel(y_true)
  float* SKH   = ws + 7LL * NVT;              // soft_skel(hard)
  float* SKP   = ws + 8LL * NVT;              // soft_skel(prob), transient
  float* accT  = ws + 8LL * NVT;              // EDT(y_true) acc (reuses SKP slot)
  float* accH  = ws + 9LL * NVT;              // EDT(hard)  acc
  float* accF  = ws + 10LL * NVT;             // scalar accumulators
  int*   accI  = (int*)accF;

  dim3 gp(PW_BLOCKS), bp(256);
  dim3 gs(NROWS),     bs(WW);

  k_init_acc<<<1, 64, 0, stream>>>(accF);
  k_prep<<<gp, bp, 0, stream>>>(net, tgt, prob, ytrue, hard, accF);

  // clDice skeletons + sums
  soft_skel_seq(prob, SKP, A, B, C, stream);
  k_dot2<<<gp, bp, 0, stream>>>(SKP, ytrue, accF, 6, 7);   // sum sp*yt, sum sp
  soft_skel_seq(ytrue, SKT, A, B, C, stream);
  k_dot2<<<gp, bp, 0, stream>>>(SKT, prob, accF, 8, 9);    // sum st*p,  sum st
  soft_skel_seq(hard, SKH, A, B, C, stream);

  // EDT approximations (SKP slot is free now -> accT)
  edt_seq(ytrue, accT, A, B, stream);
  edt_seq(hard,  accH, A, B, stream);

  // Directional (Sobel cosine) loss
  k_sobel<<<gs, bs, 0, stream>>>(net, tgt, accF);

  // Union loss: per-batch radius extrema, then weighted sums
  k_radius<<<gp, bp, 0, stream>>>(accT, SKT, ytrue, accH, SKH, prob, hard, accI);
  k_union <<<gp, bp, 0, stream>>>(accT, SKT, ytrue, accH, SKH, prob, hard, accI, accF);

  k_final<<<1, 1, 0, stream>>>(accF, out);
}